// HashLayer_5033701671492
// MI455X (gfx1250) — compile-verified
//
#include <hip/hip_runtime.h>

typedef __attribute__((ext_vector_type(2))) float v2f;
typedef __attribute__((ext_vector_type(8))) float v8f;

#define ROWS_PER_BLOCK 256
#define FEAT_STRIDE 36   // floats; 144B row stride: 16B-aligned, conflict-free strided b64 reads

__global__ __launch_bounds__(256) void hash_layer_wmma(
    const float* __restrict__ x,
    const int*   __restrict__ hashs,
    const float* __restrict__ emb,
    const float* __restrict__ Wm,      // (32 out, 32 in) row-major
    const float* __restrict__ bias,    // (32,)
    float* __restrict__ out,           // (rows, 32)
    int nrows)
{
    __shared__ int s_hash[64];
    __shared__ __align__(16) float s_feat[ROWS_PER_BLOCK * FEAT_STRIDE]; // also reused as s_out (256*32)

    const int tid   = threadIdx.x;
    const int lane  = tid & 31;
    const int wave  = tid >> 5;
    const int lrow  = lane & 15;       // M (A/D) or N (B/D) sub-index
    const int khalf = (lane >> 4) * 2; // K offset 0 or 2 within a K=4 chunk

    if (tid < 64) s_hash[tid] = hashs[tid];

    // ---- B fragments (map_weight) in registers: B[k][n] = W[n][k] ----
    v2f bfrag[2][8];
#pragma unroll
    for (int n = 0; n < 2; ++n)
#pragma unroll
        for (int c = 0; c < 8; ++c)
            bfrag[n][c] = *(const v2f*)(Wm + (size_t)(n * 16 + lrow) * 32 + c * 4 + khalf);

    const float bias_n[2] = { bias[lrow], bias[16 + lrow] };

    __syncthreads(); // s_hash visible

    // ---- Phase A: one row per thread (hash + embedding gather -> LDS) ----
    const int row    = blockIdx.x * ROWS_PER_BLOCK + tid;
    const int rclamp = row < nrows ? row : nrows - 1;
    const float4* xr = (const float4*)(x + (size_t)rclamp * 32);

    unsigned hsum = 0u;
#pragma unroll
    for (int j4 = 0; j4 < 8; ++j4) {
        float4 v = xr[j4];
        hsum += (unsigned)s_hash[(j4 * 4 + 0) * 2 + (int)rintf(v.x)];
        hsum += (unsigned)s_hash[(j4 * 4 + 1) * 2 + (int)rintf(v.y)];
        hsum += (unsigned)s_hash[(j4 * 4 + 2) * 2 + (int)rintf(v.z)];
        hsum += (unsigned)s_hash[(j4 * 4 + 3) * 2 + (int)rintf(v.w)];
    }
    unsigned idx = hsum & 0x3FFFFFu; // Python mod 2^22 on int32 == bitwise AND

    const unsigned off[4] = { 0u, 4194304u, 6291456u, 7340032u };
    float4* frow = (float4*)(s_feat + tid * FEAT_STRIDE);
#pragma unroll
    for (int i = 0; i < 4; ++i) {
        const float4* e = (const float4*)(emb + (size_t)((idx >> i) + off[i]) * 8);
        frow[i * 2 + 0] = e[0];
        frow[i * 2 + 1] = e[1];
    }
    __syncthreads(); // features visible (wave-local in fact, but keep it safe)

    // ---- Phase B: each wave = 2 row-tiles x 2 col-tiles, 8 chained K=4 WMMAs each ----
    v8f accs[2][2];
#pragma unroll
    for (int t = 0; t < 2; ++t) {
        const int rbase = wave * 32 + t * 16;
#pragma unroll
        for (int n = 0; n < 2; ++n) {
            v8f acc;
            const float bv = bias_n[n];
#pragma unroll
            for (int v = 0; v < 8; ++v) acc[v] = bv; // fold bias into C

#pragma unroll
            for (int c = 0; c < 8; ++c) {
                // A 16x4 f32 layout: lane<16 -> {K=4c,4c+1}, lane>=16 -> {K=4c+2,4c+3}, M=lrow
                v2f a = *(const v2f*)(s_feat + (size_t)(rbase + lrow) * FEAT_STRIDE + c * 4 + khalf);
                acc = __builtin_amdgcn_wmma_f32_16x16x4_f32(
                    false, a, false, bfrag[n][c], (short)0, acc, false, false);
            }
            accs[t][n] = acc;
        }
    }

    // ---- Epilogue: clip + RNE quantize, transpose through LDS, coalesced b128 stores ----
    __syncthreads(); // all A-fragment reads complete -> safe to alias s_feat as row-major s_out
    float* s_out = s_feat; // [256][32] row-major

#pragma unroll
    for (int t = 0; t < 2; ++t) {
        const int rofs = wave * 32 + t * 16 + ((lane >> 4) << 3); // D: lane<16 -> M=v, lane>=16 -> M=v+8
#pragma unroll
        for (int n = 0; n < 2; ++n) {
            const int col = n * 16 + lrow;
#pragma unroll
            for (int v = 0; v < 8; ++v) {
                float val = fminf(fmaxf(accs[t][n][v], -1.0f), 127.0f / 128.0f);
                val = rintf(val * 128.0f) * (1.0f / 128.0f); // fake_quant forward == RNE quantize
                s_out[(rofs + v) * 32 + col] = val;
            }
        }
    }
    __syncthreads();

    const size_t base = (size_t)blockIdx.x * ROWS_PER_BLOCK * 32;
    const float4* s4  = (const float4*)s_out;
    float4*       o4  = (float4*)(out + base);
    if ((blockIdx.x + 1) * ROWS_PER_BLOCK <= nrows) {
        // fast path: flat, fully coalesced 128B stores
#pragma unroll
        for (int k = 0; k < 8; ++k)
            o4[k * 256 + tid] = s4[k * 256 + tid];
    } else {
        // tail block: each float4 lies within one output row
#pragma unroll
        for (int k = 0; k < 8; ++k) {
            const int f4   = k * 256 + tid;
            const int grow = blockIdx.x * ROWS_PER_BLOCK + (f4 >> 3);
            if (grow < nrows) o4[f4] = s4[f4];
        }
    }
}

extern "C" void kernel_launch(void* const* d_in, const int* in_sizes, int n_in,
                              void* d_out, int out_size, void* d_ws, size_t ws_size,
                              hipStream_t stream) {
    const float* x    = (const float*)d_in[0]; // (BATCH, 32) f32
    const int*   hsh  = (const int*)  d_in[1]; // (32, 2) i32
    const float* emb  = (const float*)d_in[2]; // (7864320, 8) f32
    const float* Wm   = (const float*)d_in[3]; // (32, 32) f32
    const float* bias = (const float*)d_in[4]; // (32,) f32
    float*       out  = (float*)d_out;

    const int nrows  = in_sizes[0] / 32;
    const int blocks = (nrows + ROWS_PER_BLOCK - 1) / ROWS_PER_BLOCK;
    hash_layer_wmma<<<blocks, 256, 0, stream>>>(x, hsh, emb, Wm, bias, out, nrows);
}